// GraphSAGE_15324443312421
// MI455X (gfx1250) — compile-verified
//
#include <hip/hip_runtime.h>

#define NN 20000
#define NE 640000
#define INF 300
#define HF 512
#define NC 256

typedef float v2f __attribute__((ext_vector_type(2)));
typedef float v8f __attribute__((ext_vector_type(8)));

__global__ void zero_i32(int* __restrict__ p, long n) {
    long i = (long)blockIdx.x * blockDim.x + threadIdx.x;
    long stride = (long)gridDim.x * blockDim.x;
    for (; i < n; i += stride) p[i] = 0;
}

__global__ void degree_kernel(const int* __restrict__ dst, int* __restrict__ deg, int nE) {
    int e = blockIdx.x * blockDim.x + threadIdx.x;
    if (e < nE) atomicAdd(&deg[dst[e]], 1);
}

// Single-block exclusive prefix scan over n (=20000) degree counts -> CSR row offsets.
// Hillis-Steele in LDS, chunked by 1024 with a running carry. n is tiny; this is
// microseconds and keeps the pipeline on one stream with no host sync.
__global__ __launch_bounds__(1024) void scan_kernel(const int* __restrict__ deg,
                                                    int* __restrict__ off, int n) {
    __shared__ int buf[1024];
    __shared__ int carry_s;
    const int t = threadIdx.x;
    if (t == 0) carry_s = 0;
    __syncthreads();
    for (int base = 0; base < n; base += 1024) {
        const int i = base + t;
        const int v = (i < n) ? deg[i] : 0;
        buf[t] = v;
        __syncthreads();
        for (int d = 1; d < 1024; d <<= 1) {
            int add = (t >= d) ? buf[t - d] : 0;
            __syncthreads();
            buf[t] += add;
            __syncthreads();
        }
        const int carry = carry_s;
        if (i < n) off[i] = carry + buf[t] - v;   // exclusive
        const int total = buf[1023];
        __syncthreads();
        if (t == 0) carry_s = carry + total;
        __syncthreads();
    }
    if (t == 0) off[n] = carry_s;
}

__global__ void fill_csr_kernel(const int* __restrict__ src, const int* __restrict__ dst,
                                const int* __restrict__ off, int* __restrict__ cursor,
                                int* __restrict__ csr, int nE) {
    int e = blockIdx.x * blockDim.x + threadIdx.x;
    if (e < nE) {
        int d = dst[e];
        int pos = off[d] + atomicAdd(&cursor[d], 1);
        csr[pos] = src[e];
    }
}

// One wave per destination node: stream in-neighbor rows through the normal
// (non-atomic) load path, accumulate in VGPRs, write the mean once.
// Rows are contiguous (D*4 bytes) and lane-coalesced; the whole working set
// (x 24MB / h 41MB) is L2-resident on the 192MB L2.
template <int MAXJ>
__global__ void gather_mean_kernel(const float* __restrict__ x,
                                   const int* __restrict__ off,
                                   const int* __restrict__ csr,
                                   float* __restrict__ mean,
                                   int D, int nNodes) {
    const int wavesPerBlock = blockDim.x >> 5;
    const int node = blockIdx.x * wavesPerBlock + (threadIdx.x >> 5);
    const int lane = threadIdx.x & 31;
    if (node >= nNodes) return;

    const int beg = off[node];
    const int end = off[node + 1];

    float acc[MAXJ];
#pragma unroll
    for (int j = 0; j < MAXJ; ++j) acc[j] = 0.0f;

    for (int e = beg; e < end; ++e) {
        const int s = csr[e];                       // wave-uniform
        const float* __restrict__ xs = x + (size_t)s * D;
#pragma unroll
        for (int j = 0; j < MAXJ; ++j) {
            const int f = lane + 32 * j;
            if (f < D) acc[j] += xs[f];
        }
    }

    const float scale = 1.0f / fmaxf((float)(end - beg), 1.0f);
    float* __restrict__ o = mean + (size_t)node * D;
#pragma unroll
    for (int j = 0; j < MAXJ; ++j) {
        const int f = lane + 32 * j;
        if (f < D) o[f] = acc[j] * scale;
    }
}

// Fused SAGEConv GEMM: Out = act( Xs*Ws + Xn*Wn + bias )
// One wave computes a 16x16 f32 tile via V_WMMA_F32_16X16X4_F32, stepping K by 4
// and accumulating both products into one 8-VGPR accumulator. Full fp32: the
// workload is aggregation-bound, so there is no reason to drop precision.
__global__ __launch_bounds__(256) void sage_gemm_wmma(
    const float* __restrict__ Xs,   // [M,K] self features
    const float* __restrict__ Xn,   // [M,K] neighbor means
    const float* __restrict__ Ws,   // [K,N]
    const float* __restrict__ Wn,   // [K,N]
    const float* __restrict__ bias, // [N]
    float* __restrict__ Out,        // [M,N]
    int M, int K, int N, int relu)
{
    const int lane = threadIdx.x & 31;
    const int wave = threadIdx.x >> 5;
    const int m0 = blockIdx.x * 16;
    const int n0 = (blockIdx.y * 8 + wave) * 16;
    if (m0 >= M || n0 >= N) return;   // wave-uniform: EXEC stays all-ones for WMMA

    const int half = lane >> 4;       // A/B VGPR0 holds K=2h..2h+1 per half-wave
    const int ml   = lane & 15;
    const int arow = m0 + ml;

    const float* __restrict__ xsrow = Xs + (size_t)arow * K;
    const float* __restrict__ xnrow = Xn + (size_t)arow * K;

    v8f c = {};
    for (int k = 0; k < K; k += 4) {
        const int ka = k + 2 * half;
        v2f a, b;

        // self term
        a.x = xsrow[ka];
        a.y = xsrow[ka + 1];
        b.x = Ws[(size_t)ka       * N + n0 + ml];
        b.y = Ws[(size_t)(ka + 1) * N + n0 + ml];
        c = __builtin_amdgcn_wmma_f32_16x16x4_f32(
                false, a, false, b, (short)0, c, false, false);

        // neighbor term (mean already applied in gather)
        a.x = xnrow[ka];
        a.y = xnrow[ka + 1];
        b.x = Wn[(size_t)ka       * N + n0 + ml];
        b.y = Wn[(size_t)(ka + 1) * N + n0 + ml];
        c = __builtin_amdgcn_wmma_f32_16x16x4_f32(
                false, a, false, b, (short)0, c, false, false);
    }

    // C/D layout: VGPR v -> row m0 + v + 8*half, col n0 + ml
    const float bv = bias[n0 + ml];
#pragma unroll
    for (int v = 0; v < 8; ++v) {
        const int r = m0 + v + 8 * half;
        float val = c[v] + bv;
        if (relu) val = fmaxf(val, 0.0f);
        Out[(size_t)r * N + n0 + ml] = val;
    }
}

extern "C" void kernel_launch(void* const* d_in, const int* in_sizes, int n_in,
                              void* d_out, int out_size, void* d_ws, size_t ws_size,
                              hipStream_t stream) {
    const float* x   = (const float*)d_in[0];
    const float* Ws1 = (const float*)d_in[1];
    const float* Wn1 = (const float*)d_in[2];
    const float* b1  = (const float*)d_in[3];
    const float* Ws2 = (const float*)d_in[4];
    const float* Wn2 = (const float*)d_in[5];
    const float* b2  = (const float*)d_in[6];
    const int*   src = (const int*)d_in[7];
    const int*   dst = (const int*)d_in[8];
    float*       out = (float*)d_out;

    // Workspace layout: [deg|cursor|off|csr] ints, then [mean1|h|mean2] floats
    int* ip     = (int*)d_ws;
    int* deg    = ip;                         // NN
    int* cursor = deg + NN;                   // NN
    int* off    = cursor + NN;                // NN+1
    int* csr    = off + (NN + 1);             // NE
    float* fp   = (float*)(csr + NE);
    float* mean1 = fp;                        // NN*INF
    float* h     = mean1 + (size_t)NN * INF;  // NN*HF
    float* mean2 = h + (size_t)NN * HF;       // NN*HF

    const size_t need = ((size_t)(3 * NN + 1 + NE)) * sizeof(int) +
                        ((size_t)NN * INF + 2 * (size_t)NN * HF) * sizeof(float);
    if (ws_size < need) return;

    const int BLK = 256;

    // --- CSR build (shared by both layers) ---
    zero_i32<<<256, BLK, 0, stream>>>(deg, 2L * NN);            // deg + cursor contiguous
    degree_kernel<<<(NE + BLK - 1) / BLK, BLK, 0, stream>>>(dst, deg, NE);
    scan_kernel<<<1, 1024, 0, stream>>>(deg, off, NN);
    fill_csr_kernel<<<(NE + BLK - 1) / BLK, BLK, 0, stream>>>(src, dst, off, cursor, csr, NE);

    // --- Layer 1: gather mean + fused GEMM + ReLU ---
    gather_mean_kernel<10><<<(NN + 7) / 8, BLK, 0, stream>>>(x, off, csr, mean1, INF, NN);
    sage_gemm_wmma<<<dim3(NN / 16, HF / 128), BLK, 0, stream>>>(
        x, mean1, Ws1, Wn1, b1, h, NN, INF, HF, 1);

    // --- Layer 2: gather mean + fused GEMM ---
    gather_mean_kernel<16><<<(NN + 7) / 8, BLK, 0, stream>>>(h, off, csr, mean2, HF, NN);
    sage_gemm_wmma<<<dim3(NN / 16, NC / 128), BLK, 0, stream>>>(
        h, mean2, Ws2, Wn2, b2, out, NN, HF, NC, 0);
}